// shared_layers_77446850281579
// MI455X (gfx1250) — compile-verified
//
#include <hip/hip_runtime.h>
#include <hip/hip_bf16.h>

#define BATCH 16
#define NPTS  2048
#define KNN   20

typedef __attribute__((ext_vector_type(2))) float v2f;
typedef __attribute__((ext_vector_type(8))) float v8f;

__device__ inline v8f wmma4(v2f a, v2f b, v8f c) {
  // D = A(16x4 f32) * B(4x16 f32) + C(16x16 f32)  -> v_wmma_f32_16x16x4_f32
  return __builtin_amdgcn_wmma_f32_16x16x4_f32(false, a, false, b, (short)0, c, false, false);
}

__device__ inline v8f v8_splat(float x) { v8f r; for (int i = 0; i < 8; ++i) r[i] = x; return r; }

__device__ inline void atomicMaxF(float* a, float v) {
  // IEEE-ordered float max via integer atomics (works with -big init)
  if (!(__float_as_uint(v) >> 31)) atomicMax((int*)a, __float_as_int(v));
  else atomicMin((unsigned int*)a, __float_as_uint(v));
}

// ---------------- init x5 to -big ----------------
__global__ void k_init_neg(float* p, int n) {
  int i = blockIdx.x * blockDim.x + threadIdx.x;
  if (i < n) p[i] = -3.0e38f;
}

// ---------------- fold linear-linear conv pairs ----------------
__global__ void k_fuse(const float* __restrict__ w1, const float* __restrict__ b1,
                       const float* __restrict__ w2, const float* __restrict__ b2,
                       const float* __restrict__ w3, const float* __restrict__ b3,
                       const float* __restrict__ w4, const float* __restrict__ b4,
                       const float* __restrict__ w5,
                       float* wd12, float* wcd12, float* bf12,
                       float* wd34, float* wcd34, float* bf34,
                       float* wd5, float* wcd5) {
  int t = blockIdx.x * blockDim.x + threadIdx.x;
  if (t >= 64 * 64) return;
  int o = t >> 6, c = t & 63;
  // W34 = w4 @ w3 ; split [Wd | Wc], store Wd and (Wc - Wd)
  float sd = 0.f, sc = 0.f;
  for (int j = 0; j < 64; ++j) {
    float a = w4[o * 64 + j];
    sd += a * w3[j * 128 + c];
    sc += a * w3[j * 128 + 64 + c];
  }
  wd34[o * 64 + c] = sd;
  wcd34[o * 64 + c] = sc - sd;
  // W5 split directly
  float d5 = w5[o * 128 + c];
  wd5[o * 64 + c] = d5;
  wcd5[o * 64 + c] = w5[o * 128 + 64 + c] - d5;
  if (c < 4) {
    float s1 = 0.f, s2 = 0.f;
    if (c < 3) {
      for (int j = 0; j < 64; ++j) {
        float a = w2[o * 64 + j];
        s1 += a * w1[j * 6 + c];
        s2 += a * w1[j * 6 + 3 + c];
      }
    }
    wd12[o * 4 + c] = s1;
    wcd12[o * 4 + c] = s2 - s1;
  }
  if (c == 0) {
    float sb12 = 0.f, sb34 = 0.f;
    for (int j = 0; j < 64; ++j) {
      sb12 += w2[o * 64 + j] * b1[j];
      sb34 += w4[o * 64 + j] * b3[j];
    }
    bf12[o] = sb12 + b2[o];
    bf34[o] = sb34 + b4[o];
  }
}

// ---------------- per-row stable top-20 from an LDS score strip ----------------
__device__ inline void topk20_row(volatile float* row, int* outp, int lane) {
  for (int kk = 0; kk < KNN; ++kk) {
    float m = -3.0e38f;
    int mi = 0x7fffffff;
    for (int j = lane; j < NPTS; j += 32) {
      float v = row[j];
      if (v > m) { m = v; mi = j; }
    }
    for (int off = 16; off > 0; off >>= 1) {
      float om = __shfl_xor(m, off, 32);
      int omi = __shfl_xor(mi, off, 32);
      if (om > m || (om == m && omi < mi)) { m = om; mi = omi; }
    }
    if (lane == 0) {
      outp[kk] = mi;
      row[mi] = -3.0e38f;
    }
  }
}

// ---------------- kNN for C=3 input (VALU distances + LDS top-k) ----------------
__global__ __launch_bounds__(512) void k_knn3(const float* __restrict__ x, int* __restrict__ idx) {
  __shared__ float s_sc[16][NPTS];   // 128 KB (320 KB/WGP on CDNA5)
  __shared__ float s_ctr[3][16];
  int b = blockIdx.x / (NPTS / 16);
  int n0 = (blockIdx.x % (NPTS / 16)) * 16;
  int t = threadIdx.x, lane = t & 31, w = t >> 5;
  const float* xb = x + (size_t)b * 3 * NPTS;
  if (t < 48) {
    int c = t / 16, r = t % 16;
    s_ctr[c][r] = xb[(size_t)c * NPTS + n0 + r];
  }
  __syncthreads();
  for (int j = t; j < NPTS; j += 512) {
    float xj0 = xb[j], xj1 = xb[NPTS + j], xj2 = xb[2 * NPTS + j];
    for (int i = 0; i < 16; ++i) {
      float d0 = xj0 - s_ctr[0][i];
      float d1 = xj1 - s_ctr[1][i];
      float d2 = xj2 - s_ctr[2][i];
      s_sc[i][j] = -(d0 * d0 + d1 * d1 + d2 * d2);
    }
  }
  __syncthreads();
  topk20_row(&s_sc[w][0], idx + ((size_t)b * NPTS + n0 + w) * KNN, lane);
}

// ---------------- kNN for C=64 (fp32 WMMA distances + LDS top-k) ----------------
__global__ __launch_bounds__(512) void k_knn64(const float* __restrict__ xt, const float* __restrict__ sq,
                                               int* __restrict__ idx) {
  __shared__ float s_sc[16][NPTS];   // 128 KB
  __shared__ float s_ctr[16][64];
  __shared__ float s_sqi[16];
  int b = blockIdx.x / (NPTS / 16);
  int n0 = (blockIdx.x % (NPTS / 16)) * 16;
  int t = threadIdx.x, lane = t & 31, w = t >> 5;
  int half = lane >> 4, col = lane & 15;
  const float* xb = xt + (size_t)b * NPTS * 64;
  for (int e = t; e < 16 * 64; e += 512) {
    int r = e >> 6, c = e & 63;
    s_ctr[r][c] = xb[(size_t)(n0 + r) * 64 + c];
  }
  if (t < 16) s_sqi[t] = sq[(size_t)b * NPTS + n0 + t];
  __syncthreads();
  // A fragments: rows = centers, K = 64 channels in 16 steps of 4
  v2f afr[16];
  int arow = col;
  for (int s = 0; s < 16; ++s) {
    int k = 4 * s + 2 * half;
    afr[s][0] = s_ctr[arow][k];
    afr[s][1] = s_ctr[arow][k + 1];
  }
  for (int jt = w; jt < NPTS / 16; jt += 16) {
    int j0 = jt * 16;
    const float* bbase = xb + (size_t)(j0 + col) * 64 + 2 * half;
    __builtin_prefetch(xb + (size_t)(j0 + 256 + col) * 64, 0, 1);
    v8f acc = v8_splat(0.f);
    for (int s = 0; s < 16; ++s) {
      const float2 bv = *(const float2*)(bbase + 4 * s);
      v2f bf; bf[0] = bv.x; bf[1] = bv.y;
      acc = wmma4(afr[s], bf, acc);
    }
    float sqj = sq[(size_t)b * NPTS + j0 + col];
    for (int r = 0; r < 8; ++r) {
      int i = r + 8 * half;
      s_sc[i][j0 + col] = 2.f * acc[r] - s_sqi[i] - sqj;
    }
  }
  __syncthreads();
  topk20_row(&s_sc[w][0], idx + ((size_t)b * NPTS + n0 + w) * KNN, lane);
}

// ---------------- EdgeConv block 1 apply (C=3 -> 64, tiny; VALU) ----------------
__global__ __launch_bounds__(256) void k_edge1(const float* __restrict__ x, const int* __restrict__ idx,
                                               const float* __restrict__ wd, const float* __restrict__ wcd,
                                               const float* __restrict__ bb,
                                               float* __restrict__ outc, float* __restrict__ xtn) {
  int b = blockIdx.x / (NPTS / 64);
  int n0 = (blockIdx.x % (NPTS / 64)) * 64;
  int t = threadIdx.x;
  int p = n0 + (t >> 2);
  int o0 = (t & 3) * 16;
  const float* xb = x + (size_t)b * 3 * NPTS;
  float c0 = xb[p], c1 = xb[NPTS + p], c2 = xb[2 * NPTS + p];
  float ct[16], mx[16];
  for (int o = 0; o < 16; ++o) {
    int oo = o0 + o;
    ct[o] = wcd[oo * 4 + 0] * c0 + wcd[oo * 4 + 1] * c1 + wcd[oo * 4 + 2] * c2 + bb[oo];
    mx[o] = -3.0e38f;
  }
  const int* ip = idx + ((size_t)b * NPTS + p) * KNN;
  for (int kk = 0; kk < KNN; ++kk) {
    int j = ip[kk];
    float a0 = xb[j], a1 = xb[NPTS + j], a2 = xb[2 * NPTS + j];
    for (int o = 0; o < 16; ++o) {
      int oo = o0 + o;
      float v = wd[oo * 4 + 0] * a0 + wd[oo * 4 + 1] * a1 + wd[oo * 4 + 2] * a2 + ct[o];
      mx[o] = fmaxf(mx[o], v);
    }
  }
  float* ob = outc + (size_t)b * 192 * NPTS;   // channels 0..63
  for (int o = 0; o < 16; ++o) {
    ob[(size_t)(o0 + o) * NPTS + p] = mx[o];
    xtn[((size_t)b * NPTS + p) * 64 + o0 + o] = mx[o];
  }
}

// ---------------- row squared norms of transposed features ----------------
__global__ void k_sqnorm(const float* __restrict__ xt, float* __restrict__ sq) {
  int i = blockIdx.x * blockDim.x + threadIdx.x;
  if (i < BATCH * NPTS) {
    const float4* p = (const float4*)(xt + (size_t)i * 64);
    float s = 0.f;
    for (int q = 0; q < 16; ++q) {
      float4 v = p[q];
      s += v.x * v.x + v.y * v.y + v.z * v.z + v.w * v.w;
    }
    sq[i] = s;
  }
}

// ---------------- EdgeConv blocks 2/3 apply: fused gather + 64x64 WMMA GEMM + max ----------------
__global__ __launch_bounds__(128) void k_edge64(const float* __restrict__ xt, const int* __restrict__ idx,
                                                const float* __restrict__ wd, const float* __restrict__ wcd,
                                                const float* __restrict__ bb,
                                                float* __restrict__ outc, int cbase,
                                                float* __restrict__ xtn) {
  __shared__ float s_ctr[16][64];
  __shared__ float s_nbr[16][64];
  __shared__ int s_j[16][KNN];
  int b = blockIdx.x / (NPTS / 16);
  int n0 = (blockIdx.x % (NPTS / 16)) * 16;
  int t = threadIdx.x, lane = t & 31, w = t >> 5;
  int half = lane >> 4, col = lane & 15;
  const float* xb = xt + (size_t)b * NPTS * 64;
  for (int e = t; e < 16 * 64; e += 128) {
    int r = e >> 6, c = e & 63;
    s_ctr[r][c] = xb[(size_t)(n0 + r) * 64 + c];
  }
  for (int e = t; e < 16 * KNN; e += 128) {
    int r = e / KNN, kk = e % KNN;
    s_j[r][kk] = idx[((size_t)b * NPTS + n0 + r) * KNN + kk];
  }
  __syncthreads();
  int o0 = w * 16;                         // each wave owns one 16-row output subtile
  int arow = o0 + col;
  v2f ad[16], ac[16];
  for (int s = 0; s < 16; ++s) {
    int k = 4 * s + 2 * half;
    ad[s][0] = wd[(size_t)arow * 64 + k];  ad[s][1] = wd[(size_t)arow * 64 + k + 1];
    ac[s][0] = wcd[(size_t)arow * 64 + k]; ac[s][1] = wcd[(size_t)arow * 64 + k + 1];
  }
  // cterm = (Wc - Wd) @ ctr + bias   (loop-invariant over 20 neighbors)
  v8f cterm = v8_splat(0.f);
  for (int s = 0; s < 16; ++s) {
    int k = 4 * s + 2 * half;
    v2f bf; bf[0] = s_ctr[col][k]; bf[1] = s_ctr[col][k + 1];
    cterm = wmma4(ac[s], bf, cterm);
  }
  for (int r = 0; r < 8; ++r) cterm[r] += bb[o0 + r + 8 * half];
  v8f mx = v8_splat(-3.0e38f);
  for (int kk = 0; kk < KNN; ++kk) {
    __syncthreads();
    for (int e = t; e < 16 * 64; e += 128) {
      int r = e >> 6, c = e & 63;
      s_nbr[r][c] = xb[(size_t)s_j[r][kk] * 64 + c];
    }
    __syncthreads();
    v8f acc = v8_splat(0.f);
    for (int s = 0; s < 16; ++s) {
      int k = 4 * s + 2 * half;
      v2f bf; bf[0] = s_nbr[col][k]; bf[1] = s_nbr[col][k + 1];
      acc = wmma4(ad[s], bf, acc);
    }
    for (int r = 0; r < 8; ++r) mx[r] = fmaxf(mx[r], acc[r] + cterm[r]);
  }
  float* ob = outc + (size_t)b * 192 * NPTS + (size_t)cbase * NPTS;
  for (int r = 0; r < 8; ++r) {
    int o = o0 + r + 8 * half;
    float v = mx[r];
    ob[(size_t)o * NPTS + n0 + col] = v;
    xtn[((size_t)b * NPTS + n0 + col) * 64 + o] = v;
  }
}

// ---------------- conv6 (1024x192 WMMA GEMM) fused with global max-pool ----------------
__global__ __launch_bounds__(256) void k_conv6(const float* __restrict__ x123,
                                               const float* __restrict__ w6, const float* __restrict__ b6,
                                               float* __restrict__ x5) {
  int gb = blockIdx.x;
  int b = gb / (16 * 8);
  int rem = gb % (16 * 8);
  int oT = rem / 8, nT = rem % 8;
  int t = threadIdx.x, lane = t & 31, w = t >> 5;
  int half = lane >> 4, col = lane & 15;
  int o0 = oT * 64 + (w & 3) * 16;
  int arow = o0 + col;
  v2f af[48];                              // K = 192 in 48 steps of 4
  for (int s = 0; s < 48; ++s) {
    int k = 4 * s + 2 * half;
    af[s][0] = w6[(size_t)arow * 192 + k];
    af[s][1] = w6[(size_t)arow * 192 + k + 1];
  }
  const float* xb = x123 + (size_t)b * 192 * NPTS;
  v8f mx = v8_splat(-3.0e38f);
  for (int jt = (w >> 2); jt < 16; jt += 2) {
    int n0 = nT * 256 + jt * 16;
    __builtin_prefetch(xb + n0 + 32, 0, 1);
    v8f acc = v8_splat(0.f);
    for (int s = 0; s < 48; ++s) {
      int k = 4 * s + 2 * half;
      v2f bf;
      bf[0] = xb[(size_t)k * NPTS + n0 + col];
      bf[1] = xb[(size_t)(k + 1) * NPTS + n0 + col];
      acc = wmma4(af[s], bf, acc);
    }
    for (int r = 0; r < 8; ++r) mx[r] = fmaxf(mx[r], acc[r]);
  }
  // reduce max over the 16 columns within each lane-half, add bias, atomic-max
  for (int r = 0; r < 8; ++r) {
    float v = mx[r];
    for (int off = 1; off < 16; off <<= 1) v = fmaxf(v, __shfl_xor(v, off, 32));
    int o = o0 + r + 8 * half;
    v += b6[o];
    if (col == 0) atomicMaxF(&x5[(size_t)b * 1024 + o], v);
  }
}

extern "C" void kernel_launch(void* const* d_in, const int* in_sizes, int n_in,
                              void* d_out, int out_size, void* d_ws, size_t ws_size,
                              hipStream_t stream) {
  const float* x  = (const float*)d_in[0];
  const float* w1 = (const float*)d_in[1];  const float* b1 = (const float*)d_in[2];
  const float* w2 = (const float*)d_in[3];  const float* b2 = (const float*)d_in[4];
  const float* w3 = (const float*)d_in[5];  const float* b3 = (const float*)d_in[6];
  const float* w4 = (const float*)d_in[7];  const float* b4 = (const float*)d_in[8];
  const float* w5 = (const float*)d_in[9];  const float* b5 = (const float*)d_in[10];
  const float* w6 = (const float*)d_in[11]; const float* b6 = (const float*)d_in[12];

  float* out = (float*)d_out;                    // x123: [B,192,N]
  float* x5  = out + (size_t)BATCH * 192 * NPTS; // x5:   [B,1024]

  // workspace layout
  char* ws = (char*)d_ws;
  size_t off = 0;
  auto carve = [&](size_t bytes) { void* p = ws + off; off = (off + bytes + 255) & ~(size_t)255; return p; };
  int*   idx   = (int*)  carve((size_t)BATCH * NPTS * KNN * 4);
  float* xtA   = (float*)carve((size_t)BATCH * NPTS * 64 * 4);
  float* xtB   = (float*)carve((size_t)BATCH * NPTS * 64 * 4);
  float* sq    = (float*)carve((size_t)BATCH * NPTS * 4);
  float* wd12  = (float*)carve(64 * 4 * 4);
  float* wcd12 = (float*)carve(64 * 4 * 4);
  float* bf12  = (float*)carve(64 * 4);
  float* wd34  = (float*)carve(64 * 64 * 4);
  float* wcd34 = (float*)carve(64 * 64 * 4);
  float* bf34  = (float*)carve(64 * 4);
  float* wd5   = (float*)carve(64 * 64 * 4);
  float* wcd5  = (float*)carve(64 * 64 * 4);
  (void)ws_size; (void)in_sizes; (void)n_in; (void)out_size;

  k_init_neg<<<(BATCH * 1024 + 255) / 256, 256, 0, stream>>>(x5, BATCH * 1024);
  k_fuse<<<16, 256, 0, stream>>>(w1, b1, w2, b2, w3, b3, w4, b4, w5,
                                 wd12, wcd12, bf12, wd34, wcd34, bf34, wd5, wcd5);
  // Block 1
  k_knn3<<<BATCH * (NPTS / 16), 512, 0, stream>>>(x, idx);
  k_edge1<<<BATCH * (NPTS / 64), 256, 0, stream>>>(x, idx, wd12, wcd12, bf12, out, xtA);
  // Block 2
  k_sqnorm<<<(BATCH * NPTS) / 256, 256, 0, stream>>>(xtA, sq);
  k_knn64<<<BATCH * (NPTS / 16), 512, 0, stream>>>(xtA, sq, idx);
  k_edge64<<<BATCH * (NPTS / 16), 128, 0, stream>>>(xtA, idx, wd34, wcd34, bf34, out, 64, xtB);
  // Block 3
  k_sqnorm<<<(BATCH * NPTS) / 256, 256, 0, stream>>>(xtB, sq);
  k_knn64<<<BATCH * (NPTS / 16), 512, 0, stream>>>(xtB, sq, idx);
  k_edge64<<<BATCH * (NPTS / 16), 128, 0, stream>>>(xtB, idx, wd5, wcd5, b5, out, 128, xtA);
  // conv6 + global max pool
  k_conv6<<<BATCH * 16 * 8, 256, 0, stream>>>(out, w6, b6, x5);
}